// WaveletPacketTransform_74715251081560
// MI455X (gfx1250) — compile-verified
//
#include <hip/hip_runtime.h>

// 3-level Haar wavelet packet transform as D = T(16x16) * X(16x16) via
// chained V_WMMA_F32_16X16X4_F32 (K = 4*4 = 16). T = diag(M8, M8) where
// M8[g][t] = 2^-1.5 * (-1)^(b2*t0 + b1*t1 + b0*t2), b = g ^ (g>>1).
// Each 16x16 D tile = 8 bands x 32 input blocks (rows 0-7: blocks j0+n,
// rows 8-15: blocks j0+16+n).

typedef __attribute__((ext_vector_type(2))) float v2f;
typedef __attribute__((ext_vector_type(8))) float v8f;

#define WPT_S3 0.35355339059327373f  // 2^{-3/2}

__device__ __forceinline__ float haar_T(int r, int k) {
  // 16x16 block-diagonal transform matrix, gray-code reorder baked into rows.
  if ((r >> 3) != (k >> 3)) return 0.0f;
  const int g = r & 7;
  const int b = g ^ (g >> 1);  // source band
  const int t = k & 7;
  const int p = (((b >> 2) & 1) & (t & 1)) ^
                (((b >> 1) & 1) & ((t >> 1) & 1)) ^
                (((b >> 0) & 1) & ((t >> 2) & 1));
  return p ? -WPT_S3 : WPT_S3;
}

__global__ void __launch_bounds__(256)
wpt3_wmma_kernel(const float* __restrict__ x, float* __restrict__ out,
                 int total_tiles, int waves_total) {
  const int lane = threadIdx.x & 31;
  const int ln   = lane & 15;   // A: row M / B: col N / D: col N
  const int hi   = lane >> 4;   // selects K pair (A/B) and row half (D)
  const int wave = blockIdx.x * (blockDim.x >> 5) + (threadIdx.x >> 5);

  // A operand chunks: tile-invariant, built once.
  // Lane layout (32-bit A, 16x4): VGPR0 = T[ln][kb], VGPR1 = T[ln][kb+1],
  // with kb = 4q for lanes 0-15 and 4q+2 for lanes 16-31.
  v2f amat[4];
#pragma unroll
  for (int q = 0; q < 4; ++q) {
    const int kb = 4 * q + 2 * hi;
    amat[q].x = haar_T(ln, kb);
    amat[q].y = haar_T(ln, kb + 1);
  }

  for (int t = wave; t < total_tiles; t += waves_total) {
    const int c  = t >> 12;          // 4096 tiles per channel
    const int j0 = (t & 4095) << 5;  // 32 blocks of 8 samples per tile
    const float* xc = x + ((size_t)c << 20);  // channel row, N = 2^20

    // B operand chunks: Xtile[k][n] = xc[8*(j0 + n + 16*(k>>3)) + (k&7)].
    // Rows kb, kb+1 are adjacent in memory -> one b64 NT load per chunk.
    v2f bmat[4];
#pragma unroll
    for (int q = 0; q < 4; ++q) {
      const int kb   = 4 * q + 2 * hi;
      const int sec  = kb >> 3;  // 0: blocks j0+n, 1: blocks j0+16+n
      const int koff = kb & 7;   // even -> 8-byte aligned
      const v2f* p = reinterpret_cast<const v2f*>(
          xc + (((size_t)(j0 + ln + (sec << 4))) << 3) + koff);
      bmat[q] = __builtin_nontemporal_load(p);
    }

    // D = sum_q A_q(16x4) * B_q(4x16), accumulated in the WMMA chain.
    v8f acc = {};
#pragma unroll
    for (int q = 0; q < 4; ++q) {
      acc = __builtin_amdgcn_wmma_f32_16x16x4_f32(
          /*neg_a=*/false, amat[q], /*neg_b=*/false, bmat[q],
          /*c_mod=*/(short)0, acc, /*reuse_a=*/false, /*reuse_b=*/false);
    }

    // D[M][N]: VGPR r holds row r (lanes 0-15) / row r+8 (lanes 16-31).
    // Row r+8*hi -> band r, block j0 + 16*hi + ln = j0 + lane.
    float* oc = out + ((size_t)c << 20);  // 8 bands * 2^17 per channel
    const int j = j0 + lane;
#pragma unroll
    for (int r = 0; r < 8; ++r) {
      __builtin_nontemporal_store(acc[r], oc + ((size_t)r << 17) + j);
    }
  }
}

extern "C" void kernel_launch(void* const* d_in, const int* in_sizes, int n_in,
                              void* d_out, int out_size, void* d_ws,
                              size_t ws_size, hipStream_t stream) {
  (void)in_sizes; (void)n_in; (void)out_size; (void)d_ws; (void)ws_size;
  const float* x = (const float*)d_in[0];
  float* out = (float*)d_out;

  const int threads = 256;                       // 8 waves per block
  const int blocks = 2048;                       // 16384 waves total
  const int waves_total = blocks * (threads / 32);
  const int total_tiles = 64 * 4096;             // 262144 tiles, 16 per wave

  hipLaunchKernelGGL(wpt3_wmma_kernel, dim3(blocks), dim3(threads), 0, stream,
                     x, out, total_tiles, waves_total);
}